// attention_score_34256659153432
// MI455X (gfx1250) — compile-verified
//
#include <hip/hip_runtime.h>
#include <hip/hip_bf16.h>

typedef __attribute__((ext_vector_type(2))) float v2f;
typedef __attribute__((ext_vector_type(4))) float v4f;
typedef __attribute__((ext_vector_type(8))) float v8f;

#define IN_CH 256

// ---------------------------------------------------------------------------
// k1: deg = 1 (self-loop), out = 0
// ---------------------------------------------------------------------------
__global__ __launch_bounds__(256) void gcn_init_kernel(int* __restrict__ deg,
                                                       float* __restrict__ out,
                                                       int N) {
    int n = blockIdx.x * blockDim.x + threadIdx.x;
    if (n < N) {
        deg[n] = 1;      // self-loop contributes 1 to the degree
        out[n] = 0.0f;
    }
}

// ---------------------------------------------------------------------------
// k2: h = x @ W via V_WMMA_F32_16X16X4_F32 (full f32 precision).
// One wave32 per 16-row tile; 8 waves (tiles) per 256-thread block.
//
// A fragment (16x4 f32, ISA layout): lanes 0-15 hold K=k,k+1 of row M=lane;
//                                    lanes 16-31 hold K=k+2,k+3 of row M=lane-16.
// B fragment (4x16 f32): W[k..k+3] broadcast into ALL 16 columns
//   (B[r][n] = W[k+r] for every n). Low half-wave lanes carry rows K=k,k+1,
//   high half-wave lanes carry rows K=k+2,k+3 -> per-lane flat 8B load,
//   no per-lane zeroing / cndmask / divergence in the loop.
// Every column of D then equals x_tile @ W; lane 0 holds M=0..7 in acc[0..7]
// and lane 16 holds M=8..15.
// ---------------------------------------------------------------------------
__global__ __launch_bounds__(256) void gcn_gemv_wmma_kernel(
        const float* __restrict__ x, const float* __restrict__ W,
        float* __restrict__ h, int nTiles) {
    const int lane   = threadIdx.x & 31;
    const int wave   = threadIdx.x >> 5;
    const int tile   = blockIdx.x * 8 + wave;
    if (tile >= nTiles) return;             // wave-uniform: EXEC stays all-1s

    const int  rowSel  = lane & 15;
    const bool hi      = lane >= 16;
    const size_t rowBase = ((size_t)tile * 16 + rowSel) * IN_CH;
    const float* __restrict__ wp = W + (hi ? 2 : 0);   // per-half-wave K pair

    v8f acc = {};
    #pragma unroll 4
    for (int k = 0; k < IN_CH; k += 4) {
        // Each lane streams its row in contiguous 16B chunks (coalesced B128).
        v4f xa = *(const v4f*)(x + rowBase + k);
        v2f a;
        a[0] = hi ? xa[2] : xa[0];
        a[1] = hi ? xa[3] : xa[1];

        v2f b = *(const v2f*)(wp + k);      // {W[k],W[k+1]} or {W[k+2],W[k+3]}

        acc = __builtin_amdgcn_wmma_f32_16x16x4_f32(
                  /*neg_a=*/false, a, /*neg_b=*/false, b,
                  /*c_mod=*/(short)0, acc, /*reuse_a=*/false, /*reuse_b=*/false);
    }

    if (rowSel == 0) {          // lanes 0 and 16: any column of D is the result
        int base = tile * 16 + (hi ? 8 : 0);
        #pragma unroll
        for (int i = 0; i < 8; ++i) h[base + i] = acc[i];
    }
}

// ---------------------------------------------------------------------------
// k3: degree accumulation at dst (integer atomics -> bit-deterministic)
// ---------------------------------------------------------------------------
__global__ __launch_bounds__(256) void gcn_deg_kernel(
        const long long* __restrict__ ei, int E, int* __restrict__ deg) {
    int e = blockIdx.x * blockDim.x + threadIdx.x;
    if (e < E) {
        int dst = (int)ei[(size_t)E + e];
        atomicAdd(&deg[dst], 1);
    }
}

// ---------------------------------------------------------------------------
// k4: dinv = rsqrt(deg)   (deg >= 1 always, thanks to the self-loop)
// ---------------------------------------------------------------------------
__global__ __launch_bounds__(256) void gcn_dinv_kernel(
        const int* __restrict__ deg, float* __restrict__ dinv, int N) {
    int n = blockIdx.x * blockDim.x + threadIdx.x;
    if (n < N) dinv[n] = rsqrtf((float)deg[n]);
}

// ---------------------------------------------------------------------------
// k5: out[dst] += dinv[src] * dinv[dst] * h[src]
// (h and dinv are 400 KB each -> resident in the 192 MB L2; gathers are cheap)
// ---------------------------------------------------------------------------
__global__ __launch_bounds__(256) void gcn_scatter_kernel(
        const long long* __restrict__ ei, int E,
        const float* __restrict__ h, const float* __restrict__ dinv,
        float* __restrict__ out) {
    int e = blockIdx.x * blockDim.x + threadIdx.x;
    if (e < E) {
        int src = (int)ei[e];
        int dst = (int)ei[(size_t)E + e];
        float v = dinv[src] * dinv[dst] * h[src];
        atomicAdd(&out[dst], v);
    }
}

// ---------------------------------------------------------------------------
// k6: out += dinv^2 * h (self-loop message) + bias
// ---------------------------------------------------------------------------
__global__ __launch_bounds__(256) void gcn_final_kernel(
        float* __restrict__ out, const float* __restrict__ h,
        const float* __restrict__ dinv, const float* __restrict__ b, int N) {
    int n = blockIdx.x * blockDim.x + threadIdx.x;
    if (n < N) out[n] = out[n] + dinv[n] * dinv[n] * h[n] + b[0];
}

// ---------------------------------------------------------------------------
extern "C" void kernel_launch(void* const* d_in, const int* in_sizes, int n_in,
                              void* d_out, int out_size, void* d_ws, size_t ws_size,
                              hipStream_t stream) {
    const float*     x  = (const float*)d_in[0];      // [N, 256] f32
    const long long* ei = (const long long*)d_in[1];  // [2, E] int64
    const float*     W  = (const float*)d_in[2];      // [256, 1] f32
    const float*     b  = (const float*)d_in[3];      // [1] f32

    const int N = in_sizes[0] / IN_CH;   // 100000 (multiple of 16)
    const int E = in_sizes[1] / 2;       // 3200000
    float* out = (float*)d_out;

    // workspace: h [N f32] | dinv [N f32] | deg [N i32]
    float* h    = (float*)d_ws;
    float* dinv = h + N;
    int*   deg  = (int*)(dinv + N);

    const int tpb = 256;
    const int nTiles = N / 16;

    gcn_init_kernel<<<(N + tpb - 1) / tpb, tpb, 0, stream>>>(deg, out, N);
    gcn_gemv_wmma_kernel<<<(nTiles + 7) / 8, tpb, 0, stream>>>(x, W, h, nTiles);
    gcn_deg_kernel<<<(E + tpb - 1) / tpb, tpb, 0, stream>>>(ei, E, deg);
    gcn_dinv_kernel<<<(N + tpb - 1) / tpb, tpb, 0, stream>>>(deg, dinv, N);
    gcn_scatter_kernel<<<(E + tpb - 1) / tpb, tpb, 0, stream>>>(ei, E, h, dinv, out);
    gcn_final_kernel<<<(N + tpb - 1) / tpb, tpb, 0, stream>>>(out, h, dinv, b, N);
}